// StrokeModificationModule_71322226917571
// MI455X (gfx1250) — compile-verified
//
#include <hip/hip_runtime.h>
#include <math.h>

// Problem constants (match reference)
#define B       32
#define P       4096
#define S       128
#define D_INT   512
#define D_VC    512
#define D_PATCH 768
#define H_      4
#define D_CONCAT 1024   // D_INT + D_VC
#define D_HEAD   256    // D_CONCAT / H

typedef __attribute__((ext_vector_type(2))) float v2f;
typedef __attribute__((ext_vector_type(8))) float v8f;

// ---------------------------------------------------------------------------
// GEMM1: q_flat[32 x 1024] = concat(h_int,h_vc)[32 x 1024] @ B + qb_flat
//        B[k, n] = qW_flat[n*1024 + k]   (n = h*256+d, k = c)
// WMMA f32 16x16x4, 2 M-tiles x 64 N-tiles = 128 waves, K loop of 256 steps.
// Lane layout (ISA 7.12.2): A/B component j <-> K = 4*step + j + 2*(lane>=16);
// D VGPR r <-> row (16*mtile + r + 8*(lane>=16)), col = 16*ntile + (lane&15).
// ---------------------------------------------------------------------------
__global__ void qproj_kernel(const float* __restrict__ h_int,
                             const float* __restrict__ h_vc,
                             const float* __restrict__ qW,
                             const float* __restrict__ qb,
                             float* __restrict__ q_flat) {
    const int wave  = (blockIdx.x * blockDim.x + threadIdx.x) >> 5;  // 0..127
    const int lane  = threadIdx.x & 31;
    const int mtile = wave >> 6;       // 0..1
    const int ntile = wave & 63;       // 0..63
    const int lo = lane & 15, hi = lane >> 4;
    const int m = mtile * 16 + lo;     // A row this lane supplies
    const int n = ntile * 16 + lo;     // B/D column this lane supplies

    const float* aRow0 = h_int + (size_t)m * D_INT;
    const float* aRow1 = h_vc  + (size_t)m * D_VC;
    const float* bCol  = qW + (size_t)n * D_CONCAT;   // contiguous in k

    v8f acc = {};
    for (int s = 0; s < D_CONCAT / 4; ++s) {
        const int k0 = 4 * s + 2 * hi;
        v2f a = *(const v2f*)((k0 < D_INT) ? (aRow0 + k0) : (aRow1 + (k0 - D_INT)));
        v2f b = *(const v2f*)(bCol + k0);
        acc = __builtin_amdgcn_wmma_f32_16x16x4_f32(
                  false, a, false, b, (short)0, acc, false, false);
    }
    const float bias = qb[n];
#pragma unroll
    for (int r = 0; r < 8; ++r) {
        const int row = mtile * 16 + r + 8 * hi;       // < 32
        q_flat[(size_t)row * D_CONCAT + n] = acc[r] + bias;
    }
}

// ---------------------------------------------------------------------------
// GEMM2: wvec[32 x 768] = 0.25 * q_flat[32 x 1024] @ B,  B[k,n] = kW_flat[k*768+n]
// 2 M-tiles x 48 N-tiles = 96 waves.
// ---------------------------------------------------------------------------
__global__ void wproj_kernel(const float* __restrict__ q_flat,
                             const float* __restrict__ kW,
                             float* __restrict__ wvec) {
    const int wave  = (blockIdx.x * blockDim.x + threadIdx.x) >> 5;  // 0..95
    const int lane  = threadIdx.x & 31;
    const int mtile = wave / 48;       // 0..1
    const int ntile = wave % 48;       // 0..47
    const int lo = lane & 15, hi = lane >> 4;
    const int m = mtile * 16 + lo;
    const int n = ntile * 16 + lo;

    const float* aRow  = q_flat + (size_t)m * D_CONCAT;
    const float* bBase = kW + n;       // stride D_PATCH in k

    v8f acc = {};
    for (int s = 0; s < D_CONCAT / 4; ++s) {
        const int k0 = 4 * s + 2 * hi;
        v2f a = *(const v2f*)(aRow + k0);
        v2f b;
        b.x = bBase[(size_t)k0 * D_PATCH];
        b.y = bBase[(size_t)(k0 + 1) * D_PATCH];
        acc = __builtin_amdgcn_wmma_f32_16x16x4_f32(
                  false, a, false, b, (short)0, acc, false, false);
    }
#pragma unroll
    for (int r = 0; r < 8; ++r) {
        const int row = mtile * 16 + r + 8 * hi;
        wvec[(size_t)row * D_PATCH + n] = 0.25f * acc[r];
    }
}

// ---------------------------------------------------------------------------
// bias[b] = 0.25 * sum_k q_flat[b,k] * kb_flat[k]     (tiny, one block)
// ---------------------------------------------------------------------------
__global__ void bias_kernel(const float* __restrict__ q_flat,
                            const float* __restrict__ kb,
                            float* __restrict__ bias) {
    __shared__ float sh[256];
    const int t = threadIdx.x;
    const int b = t >> 3, j = t & 7;        // 8 threads per b, 128 k each
    const float* q = q_flat + (size_t)b * D_CONCAT + j * 128;
    const float* c = kb + j * 128;
    float p = 0.f;
    for (int k = 0; k < 128; ++k) p += q[k] * c[k];
    sh[t] = p;
    __syncthreads();
    if (j == 0) {
        float s = 0.f;
#pragma unroll
        for (int i = 0; i < 8; ++i) s += sh[b * 8 + i];
        bias[b] = 0.25f * s;
    }
}

// ---------------------------------------------------------------------------
// prep: seg[p] = searchsorted(boundaries, p, 'right') - 1, dump->128; zero ssum
// ---------------------------------------------------------------------------
__global__ void prep_kernel(const int* __restrict__ boundaries,
                            int* __restrict__ seg,
                            float* __restrict__ ssum) {
    const int idx = blockIdx.x * blockDim.x + threadIdx.x;
    if (idx < P) {
        const int p = idx;
        int lo = 0, hi = S + 1;             // 129 boundaries
        while (lo < hi) {
            int mid = (lo + hi) >> 1;
            if (boundaries[mid] <= p) lo = mid + 1; else hi = mid;
        }
        int sg = lo - 1;
        if (sg < 0 || sg >= S) sg = S;      // dump bucket
        seg[p] = sg;
    } else if (idx < P + B * (S + 1)) {
        ssum[idx - P] = 0.f;
    }
}

// ---------------------------------------------------------------------------
// Streaming pass: ps[b,p] = PT[b,p,:].w[b] + bias[b]; accumulate into ssum[b,seg].
// Wave-per-patch, float4 coalesced loads (512 B / instruction), butterfly
// reduce, run-length-compressed f32 atomics. Grid: 32 b x 16 chunks; each of
// 8 waves handles 32 contiguous patches.
// ---------------------------------------------------------------------------
__global__ void score_kernel(const float* __restrict__ pt,
                             const float* __restrict__ wvec,
                             const float* __restrict__ bias,
                             const int* __restrict__ seg,
                             float* __restrict__ ssum) {
    const int b     = blockIdx.x >> 4;      // 0..31
    const int chunk = blockIdx.x & 15;      // 0..15
    const int wave  = threadIdx.x >> 5;     // 0..7
    const int lane  = threadIdx.x & 31;

    const float4* w4 = (const float4*)(wvec + (size_t)b * D_PATCH);
    float4 wr[6];
#pragma unroll
    for (int i = 0; i < 6; ++i) wr[i] = w4[i * 32 + lane];
    const float bb = bias[b];

    const float4* pt4 = (const float4*)(pt + (size_t)b * P * D_PATCH);
    const int p0 = chunk * 256 + wave * 32;

    int   curSeg = seg[p0];
    float curAcc = 0.f;
    for (int t = 0; t < 32; ++t) {
        const int p = p0 + t;
        const float4* row = pt4 + (size_t)p * (D_PATCH / 4);
        float acc = 0.f;
#pragma unroll
        for (int i = 0; i < 6; ++i) {
            float4 v = row[i * 32 + lane];
            acc += v.x * wr[i].x + v.y * wr[i].y + v.z * wr[i].z + v.w * wr[i].w;
        }
#pragma unroll
        for (int off = 16; off > 0; off >>= 1)
            acc += __shfl_xor(acc, off, 32);
        if (lane == 0) {
            const float ps = acc + bb;
            const int sg = seg[p];
            if (sg == curSeg) {
                curAcc += ps;
            } else {
                atomicAdd(&ssum[b * (S + 1) + curSeg], curAcc);
                curSeg = sg;
                curAcc = ps;
            }
        }
    }
    if (lane == 0) atomicAdd(&ssum[b * (S + 1) + curSeg], curAcc);
}

// ---------------------------------------------------------------------------
// finalize: out[b,s] = sigmoid( ssum[b,s] / max(cnt_s, 1) )
// cnt_s = boundaries[s+1]-boundaries[s] (exact for sorted boundaries).
// ---------------------------------------------------------------------------
__global__ void finalize_kernel(const float* __restrict__ ssum,
                                const int* __restrict__ boundaries,
                                float* __restrict__ out) {
    const int idx = blockIdx.x * blockDim.x + threadIdx.x;   // b*128 + s
    if (idx >= B * S) return;
    const int b = idx >> 7, s = idx & (S - 1);
    const float cnt   = (float)(boundaries[s + 1] - boundaries[s]);
    const float denom = fmaxf(cnt, 1.0f);
    const float x = ssum[b * (S + 1) + s] / denom;
    out[idx] = 1.0f / (1.0f + expf(-x));
}

// ---------------------------------------------------------------------------
extern "C" void kernel_launch(void* const* d_in, const int* in_sizes, int n_in,
                              void* d_out, int out_size, void* d_ws, size_t ws_size,
                              hipStream_t stream) {
    const float* h_int = (const float*)d_in[0];   // [32,512]
    const float* h_vc  = (const float*)d_in[1];   // [32,512]
    const float* pt    = (const float*)d_in[2];   // [32,4096,768]
    const int*   bnd   = (const int*)  d_in[3];   // [129]
    const float* qW    = (const float*)d_in[4];   // [4,256,1024]
    const float* qb    = (const float*)d_in[5];   // [4,256]
    const float* kW    = (const float*)d_in[6];   // [4,256,768]
    const float* kb    = (const float*)d_in[7];   // [4,256]
    float* out = (float*)d_out;                   // [32,128]

    // Workspace layout (all f32/int32, 8B-aligned where needed)
    char* ws = (char*)d_ws;
    float* q_flat = (float*)(ws);                               // 32*1024
    float* wvec   = (float*)(ws + 131072);                      // 32*768
    float* bias   = (float*)(ws + 131072 + 98304);              // 32
    int*   seg    = (int*)  (ws + 131072 + 98304 + 128);        // 4096
    float* ssum   = (float*)(ws + 131072 + 98304 + 128 + 16384);// 32*129

    // GEMM1: 128 waves (v_wmma_f32_16x16x4_f32)
    qproj_kernel<<<32, 128, 0, stream>>>(h_int, h_vc, qW, qb, q_flat);
    // GEMM2: 96 waves
    wproj_kernel<<<24, 128, 0, stream>>>(q_flat, kW, wvec);
    // bias
    bias_kernel<<<1, 256, 0, stream>>>(q_flat, kb, bias);
    // seg map + zero ssum (4096 + 4128 slots)
    prep_kernel<<<(P + B * (S + 1) + 255) / 256, 256, 0, stream>>>(bnd, seg, ssum);
    // 402 MB streaming pass (the HBM-bound part)
    score_kernel<<<B * 16, 256, 0, stream>>>(pt, wvec, bias, seg, ssum);
    // sigmoid of segment means
    finalize_kernel<<<(B * S + 255) / 256, 256, 0, stream>>>(ssum, bnd, out);
}